// GCN_23218593202704
// MI455X (gfx1250) — compile-verified
//
#include <hip/hip_runtime.h>
#include <hip/hip_bf16.h>

// ---------------------------------------------------------------------------
// GCN (3-layer, DGL GraphConv norm='both') for MI455X / gfx1250.
//   spmm(h) = D_in^-1/2 A D_out^-1/2 h  via CSR gather (built once, reused).
//   GEMMs via V_WMMA_F32_16X16X4_F32 (full f32 precision, 16x16 tile/wave).
//   Layer 2 uses spmm(h)@W2 == spmm((ds_out*h)@W2) to shrink edge traffic.
// ---------------------------------------------------------------------------

static constexpr int NN = 100000;   // nodes
static constexpr int EE = 1600000;  // edges
static constexpr int FF = 128;      // feature dim

typedef __attribute__((ext_vector_type(2))) float v2f;
typedef __attribute__((ext_vector_type(8))) float v8f;

// ------------------------------- utility -----------------------------------

__global__ void k_zero_u32(unsigned* __restrict__ p, int n) {
  int i = blockIdx.x * 256 + threadIdx.x;
  if (i < n) p[i] = 0u;
}

// ------------------------- graph preprocessing -----------------------------

__global__ void k_deg(const int* __restrict__ src, const int* __restrict__ dst,
                      unsigned* __restrict__ outdeg, unsigned* __restrict__ indeg) {
  int e = blockIdx.x * 256 + threadIdx.x;
  if (e < EE) {
    atomicAdd(&outdeg[src[e]], 1u);
    atomicAdd(&indeg[dst[e]], 1u);
  }
}

__global__ void k_ds(const unsigned* __restrict__ outdeg, const unsigned* __restrict__ indeg,
                     float* __restrict__ dsout, float* __restrict__ dsin) {
  int i = blockIdx.x * 256 + threadIdx.x;
  if (i < NN) {
    unsigned od = outdeg[i], id = indeg[i];
    dsout[i] = od ? rsqrtf((float)od) : 1.0f;
    dsin[i]  = id ? rsqrtf((float)id) : 1.0f;
  }
}

// Single-block exclusive scan of in-degrees -> CSR row offsets (N+1 entries).
__global__ void k_scan(const unsigned* __restrict__ deg, unsigned* __restrict__ offs) {
  __shared__ unsigned ts[1024];
  const int t = threadIdx.x;
  const int per = (NN + 1023) / 1024;
  int s = t * per; if (s > NN) s = NN;
  int e = s + per; if (e > NN) e = NN;
  unsigned sum = 0;
  for (int i = s; i < e; ++i) sum += deg[i];
  ts[t] = sum;
  __syncthreads();
  for (int off = 1; off < 1024; off <<= 1) {
    unsigned v = (t >= off) ? ts[t - off] : 0u;
    __syncthreads();
    if (t >= off) ts[t] += v;
    __syncthreads();
  }
  unsigned run = (t == 0) ? 0u : ts[t - 1];
  for (int i = s; i < e; ++i) { offs[i] = run; run += deg[i]; }
  if (t == 1023) offs[NN] = ts[1023];
}

__global__ void k_copy_u32(const unsigned* __restrict__ a, unsigned* __restrict__ b, int n) {
  int i = blockIdx.x * 256 + threadIdx.x;
  if (i < n) b[i] = a[i];
}

// Bucket edges by dst: esrc[] holds src ids grouped per dst row (CSR).
__global__ void k_fill(const int* __restrict__ src, const int* __restrict__ dst,
                       unsigned* __restrict__ cursor, unsigned* __restrict__ esrc) {
  int e = blockIdx.x * 256 + threadIdx.x;
  if (e < EE) {
    unsigned pos = atomicAdd(&cursor[dst[e]], 1u);
    esrc[pos] = (unsigned)src[e];
  }
}

// --------------------------- SpMM (CSR gather) -----------------------------

// agg[n,:] = dsin[n] * sum_{s in in(n)} dsout[s] * h[s,:]   (128 features)
// One wave per dst node; float4 per lane covers the 128-float row.
__global__ void k_agg128(const float* __restrict__ h, const unsigned* __restrict__ offs,
                         const unsigned* __restrict__ esrc,
                         const float* __restrict__ dsout, const float* __restrict__ dsin,
                         float* __restrict__ agg) {
  const int wv = threadIdx.x >> 5, lane = threadIdx.x & 31;
  const int n = blockIdx.x * 8 + wv;  // 12500 blocks * 8 waves == NN exactly
  const unsigned b = offs[n], e = offs[n + 1];
  float4 acc = make_float4(0.f, 0.f, 0.f, 0.f);
  for (unsigned i = b; i < e; ++i) {
    const unsigned s = esrc[i];
    const float w = dsout[s];
    const float4 v = *(const float4*)(h + (size_t)s * FF + lane * 4);
    acc.x = fmaf(w, v.x, acc.x);
    acc.y = fmaf(w, v.y, acc.y);
    acc.z = fmaf(w, v.z, acc.z);
    acc.w = fmaf(w, v.w, acc.w);
  }
  const float c = dsin[n];
  *(float4*)(agg + (size_t)n * FF + lane * 4) =
      make_float4(c * acc.x, c * acc.y, c * acc.z, c * acc.w);
}

// 64-wide variant for layer-2 (features padded 40->64). dsout already folded in.
__global__ void k_agg64(const float* __restrict__ P, const unsigned* __restrict__ offs,
                        const unsigned* __restrict__ esrc, const float* __restrict__ dsin,
                        float* __restrict__ R) {
  const int wv = threadIdx.x >> 5, lane = threadIdx.x & 31;
  const int n = blockIdx.x * 8 + wv;
  const unsigned b = offs[n], e = offs[n + 1];
  float2 acc = make_float2(0.f, 0.f);
  for (unsigned i = b; i < e; ++i) {
    const unsigned s = esrc[i];
    const float2 v = *(const float2*)(P + (size_t)s * 64 + lane * 2);
    acc.x += v.x;
    acc.y += v.y;
  }
  const float c = dsin[n];
  *(float2*)(R + (size_t)n * 64 + lane * 2) = make_float2(c * acc.x, c * acc.y);
}

// ----------------------------- WMMA GEMMs ----------------------------------

// Z[N,128] = Agg[N,128] @ W[128,128] + H[N,128] @ L[128,128]
// One 16x16 output tile per wave; 8 waves/block cover all 128 columns.
// A-frag (16x4 f32): lane l holds A[l%16][k0+2*(l/16) .. +1] in 2 VGPRs.
__global__ void k_gemm_dual(const float* __restrict__ Agg, const float* __restrict__ H,
                            const float* __restrict__ W, const float* __restrict__ L,
                            float* __restrict__ Z) {
  const int wv = threadIdx.x >> 5;       // 0..7 -> column tile
  const int lane = threadIdx.x & 31;
  const int t = lane & 15;               // A row within tile / B-D column within tile
  const int hh = lane >> 4;              // half-wave
  const int r0 = blockIdx.x << 4;        // 16 rows per block (6250 blocks)
  const int cb = wv << 4;
  const float* arow = Agg + (size_t)(r0 + t) * FF;
  const float* hrow = H + (size_t)(r0 + t) * FF;
  v8f acc = {};
#pragma unroll 4
  for (int k0 = 0; k0 < 128; k0 += 4) {
    const int kk = k0 + 2 * hh;
    v2f a1 = *(const v2f*)(arow + kk);
    v2f b1; b1.x = W[kk * FF + cb + t]; b1.y = W[(kk + 1) * FF + cb + t];
    acc = __builtin_amdgcn_wmma_f32_16x16x4_f32(false, a1, false, b1, (short)0, acc, false, false);
    v2f a2 = *(const v2f*)(hrow + kk);
    v2f b2; b2.x = L[kk * FF + cb + t]; b2.y = L[(kk + 1) * FF + cb + t];
    acc = __builtin_amdgcn_wmma_f32_16x16x4_f32(false, a2, false, b2, (short)0, acc, false, false);
  }
#pragma unroll
  for (int v = 0; v < 8; ++v)
    Z[(size_t)(r0 + v + 8 * hh) * FF + cb + t] = acc[v];
}

// P[N,64] = (dsout * H)[N,128] @ Wp[128,64]   (row scaling fused into A load)
__global__ void k_gemm_p(const float* __restrict__ H, const float* __restrict__ dsout,
                         const float* __restrict__ Wp, float* __restrict__ P) {
  const int wv = threadIdx.x >> 5;  // 0..3
  const int lane = threadIdx.x & 31;
  const int t = lane & 15, hh = lane >> 4;
  const int r0 = blockIdx.x << 4;
  const int cb = wv << 4;
  const float d = dsout[r0 + t];
  const float* hrow = H + (size_t)(r0 + t) * FF;
  v8f acc = {};
#pragma unroll 4
  for (int k0 = 0; k0 < 128; k0 += 4) {
    const int kk = k0 + 2 * hh;
    v2f a; a.x = d * hrow[kk]; a.y = d * hrow[kk + 1];
    v2f b; b.x = Wp[kk * 64 + cb + t]; b.y = Wp[(kk + 1) * 64 + cb + t];
    acc = __builtin_amdgcn_wmma_f32_16x16x4_f32(false, a, false, b, (short)0, acc, false, false);
  }
#pragma unroll
  for (int v = 0; v < 8; ++v)
    P[(size_t)(r0 + v + 8 * hh) * 64 + cb + t] = acc[v];
}

// out[N,40] = H[N,128] @ Lp[128,64] + R[N,64] + b2[40]   (cols >= 40 dropped)
__global__ void k_gemm_out(const float* __restrict__ H, const float* __restrict__ Lp,
                           const float* __restrict__ R, const float* __restrict__ b2,
                           float* __restrict__ out) {
  const int wv = threadIdx.x >> 5;  // 0..3
  if (wv == 3) return;              // cols 48..63 entirely padding
  const int lane = threadIdx.x & 31;
  const int t = lane & 15, hh = lane >> 4;
  const int r0 = blockIdx.x << 4;
  const int cb = wv << 4;
  const float* hrow = H + (size_t)(r0 + t) * FF;
  v8f acc = {};
#pragma unroll 4
  for (int k0 = 0; k0 < 128; k0 += 4) {
    const int kk = k0 + 2 * hh;
    v2f a = *(const v2f*)(hrow + kk);
    v2f b; b.x = Lp[kk * 64 + cb + t]; b.y = Lp[(kk + 1) * 64 + cb + t];
    acc = __builtin_amdgcn_wmma_f32_16x16x4_f32(false, a, false, b, (short)0, acc, false, false);
  }
  const int col = cb + t;
  if (col < 40) {
    const float bias = b2[col];
#pragma unroll
    for (int v = 0; v < 8; ++v) {
      const int row = r0 + v + 8 * hh;
      out[(size_t)row * 40 + col] = acc[v] + R[(size_t)row * 64 + col] + bias;
    }
  }
}

// ------------------------------ BatchNorm ----------------------------------

__global__ void k_colstats(const float* __restrict__ z, float* __restrict__ cs,
                           float* __restrict__ cs2) {
  const int c = threadIdx.x;  // 0..127
  const size_t r0 = (size_t)blockIdx.x * 250;  // 400 blocks * 250 rows == NN
  float s = 0.f, s2 = 0.f;
  for (int i = 0; i < 250; ++i) {
    const float v = z[(r0 + i) * FF + c];
    s += v;
    s2 = fmaf(v, v, s2);
  }
  atomicAdd(&cs[c], s);
  atomicAdd(&cs2[c], s2);
}

__global__ void k_bnparams(const float* __restrict__ cs, const float* __restrict__ cs2,
                           const float* __restrict__ g, const float* __restrict__ be,
                           float* __restrict__ bnA, float* __restrict__ bnB) {
  const int c = threadIdx.x;
  const float inv = 1.0f / (float)NN;
  const float mu = cs[c] * inv;
  const float var = cs2[c] * inv - mu * mu;
  const float a = g[c] * rsqrtf(var + 1e-5f);
  bnA[c] = a;
  bnB[c] = be[c] - mu * a;
}

__global__ void k_bnapply(const float* __restrict__ z, const float* __restrict__ bnA,
                          const float* __restrict__ bnB, float* __restrict__ hout) {
  const size_t i = (size_t)blockIdx.x * 256 + threadIdx.x;
  const int c = (int)(i & 127);
  const float v = fmaf(z[i], bnA[c], bnB[c]);
  hout[i] = v > 0.f ? v : 0.f;
}

// Pad W2/L2 [128,40] -> [128,64] with zero columns.
__global__ void k_padw(const float* __restrict__ W2, const float* __restrict__ L2,
                       float* __restrict__ Wp, float* __restrict__ Lp) {
  const int i = blockIdx.x * 256 + threadIdx.x;  // < 8192
  const int k = i >> 6, c = i & 63;
  const float w = (c < 40) ? W2[k * 40 + c] : 0.f;
  const float l = (c < 40) ? L2[k * 40 + c] : 0.f;
  Wp[i] = w;
  Lp[i] = l;
}

// ------------------------------- launcher ----------------------------------

extern "C" void kernel_launch(void* const* d_in, const int* in_sizes, int n_in,
                              void* d_out, int out_size, void* d_ws, size_t ws_size,
                              hipStream_t stream) {
  (void)in_sizes; (void)n_in; (void)out_size; (void)ws_size;

  const float* feat = (const float*)d_in[0];
  const int* src   = (const int*)d_in[1];
  const int* dst   = (const int*)d_in[2];
  const float* W0  = (const float*)d_in[3];
  const float* L0  = (const float*)d_in[4];
  const float* W1  = (const float*)d_in[5];
  const float* L1  = (const float*)d_in[6];
  const float* W2  = (const float*)d_in[7];
  const float* L2  = (const float*)d_in[8];
  const float* b2  = (const float*)d_in[9];
  const float* g0  = (const float*)d_in[10];
  const float* be0 = (const float*)d_in[11];
  const float* g1  = (const float*)d_in[12];
  const float* be1 = (const float*)d_in[13];
  float* out = (float*)d_out;

  // ---- workspace carve-out (256B aligned regions) ----
  char* base = (char*)d_ws;
  size_t off = 0;
  auto take = [&](size_t bytes) -> char* {
    char* p = base + off;
    off = (off + bytes + 255) & ~(size_t)255;
    return p;
  };
  unsigned* indeg  = (unsigned*)take((size_t)NN * 4);
  unsigned* outdeg = (unsigned*)take((size_t)NN * 4);
  unsigned* offs   = (unsigned*)take((size_t)(NN + 1) * 4);
  unsigned* cursor = (unsigned*)take((size_t)NN * 4);
  unsigned* esrc   = (unsigned*)take((size_t)EE * 4);
  float* dsout = (float*)take((size_t)NN * 4);
  float* dsin  = (float*)take((size_t)NN * 4);
  float* cs    = (float*)take(128 * 4);
  float* cs2   = (float*)take(128 * 4);
  float* bnA   = (float*)take(128 * 4);
  float* bnB   = (float*)take(128 * 4);
  float* Wp    = (float*)take(128 * 64 * 4);
  float* Lp    = (float*)take(128 * 64 * 4);
  float* bufA  = (float*)take((size_t)NN * FF * 4);  // agg, later P[N,64]
  float* bufB  = (float*)take((size_t)NN * FF * 4);  // z,   later R[N,64]
  float* bufC  = (float*)take((size_t)NN * FF * 4);  // h1, then h2

  const int BEdge = (EE + 255) / 256;   // 6250
  const int BNode = (NN + 255) / 256;   // 391
  const int BWave = NN / 8;             // 12500 (8 waves/block, 1 node/wave)
  const int BTile = NN / 16;            // 6250 (16 rows/block)
  const int BApply = (NN * FF) / 256;   // 50000

  // ---- graph preprocessing (degrees, norms, CSR) ----
  k_zero_u32<<<BNode, 256, 0, stream>>>(indeg, NN);
  k_zero_u32<<<BNode, 256, 0, stream>>>(outdeg, NN);
  k_deg<<<BEdge, 256, 0, stream>>>(src, dst, outdeg, indeg);
  k_ds<<<BNode, 256, 0, stream>>>(outdeg, indeg, dsout, dsin);
  k_scan<<<1, 1024, 0, stream>>>(indeg, offs);
  k_copy_u32<<<BNode, 256, 0, stream>>>(offs, cursor, NN);
  k_fill<<<BEdge, 256, 0, stream>>>(src, dst, cursor, esrc);
  k_padw<<<32, 256, 0, stream>>>(W2, L2, Wp, Lp);

  // ---- layer 0: h1 = relu(bn(spmm(feat)@W0 + feat@L0)) ----
  k_agg128<<<BWave, 256, 0, stream>>>(feat, offs, esrc, dsout, dsin, bufA);
  k_gemm_dual<<<BTile, 256, 0, stream>>>(bufA, feat, W0, L0, bufB);
  k_zero_u32<<<1, 256, 0, stream>>>((unsigned*)cs, 256);  // zeros cs & cs2
  k_colstats<<<400, 128, 0, stream>>>(bufB, cs, cs2);
  k_bnparams<<<1, 128, 0, stream>>>(cs, cs2, g0, be0, bnA, bnB);
  k_bnapply<<<BApply, 256, 0, stream>>>(bufB, bnA, bnB, bufC);

  // ---- layer 1: h2 = relu(bn(spmm(h1)@W1 + h1@L1)) ----
  k_agg128<<<BWave, 256, 0, stream>>>(bufC, offs, esrc, dsout, dsin, bufA);
  k_gemm_dual<<<BTile, 256, 0, stream>>>(bufA, bufC, W1, L1, bufB);
  k_zero_u32<<<1, 256, 0, stream>>>((unsigned*)cs, 256);
  k_colstats<<<400, 128, 0, stream>>>(bufB, cs, cs2);
  k_bnparams<<<1, 128, 0, stream>>>(cs, cs2, g1, be1, bnA, bnB);
  k_bnapply<<<BApply, 256, 0, stream>>>(bufB, bnA, bnB, bufC);  // h2 (h1 dead)

  // ---- layer 2: out = spmm(h2)@W2 + b2 + h2@L2 ----
  //   P = (dsout*h2)@W2p ; R = dsin * (A @ P) ; out = h2@L2p + R + b2
  k_gemm_p<<<BTile, 128, 0, stream>>>(bufC, dsout, Wp, bufA);
  k_agg64<<<BWave, 256, 0, stream>>>(bufA, offs, esrc, dsin, bufB);
  k_gemm_out<<<BTile, 128, 0, stream>>>(bufC, Lp, bufB, b2, out);
}